// SparseAttention_70231305225035
// MI455X (gfx1250) — compile-verified
//
#include <hip/hip_runtime.h>
#include <cmath>

typedef __attribute__((ext_vector_type(2))) float v2f;
typedef __attribute__((ext_vector_type(4))) float v4f;
typedef __attribute__((ext_vector_type(8))) float v8f;

#define BB 2
#define HH 8
#define EE 64

// Order-preserving uint encoding of float for atomicMax (handles negatives).
// enc(-inf-ish) region maps near 0, so init value 0u acts as -infinity.
__device__ __forceinline__ unsigned enc_ord(float f) {
  unsigned u = __float_as_uint(f);
  return (u & 0x80000000u) ? ~u : (u | 0x80000000u);
}
__device__ __forceinline__ float dec_ord(unsigned k) {
  unsigned u = (k & 0x80000000u) ? (k ^ 0x80000000u) : ~k;
  return __uint_as_float(u);
}

// ---------------------------------------------------------------------------
// Pass 0: zero output, init segment max (encoded) and segment sum
// ---------------------------------------------------------------------------
__global__ void sa_init(float* __restrict__ out, long long nOut,
                        unsigned* __restrict__ segmax,
                        float* __restrict__ segsum, long long nSeg) {
  long long i = (long long)blockIdx.x * blockDim.x + threadIdx.x;
  if (i < nOut) out[i] = 0.0f;
  if (i < nSeg) { segmax[i] = 0u; segsum[i] = 0.0f; }
}

// ---------------------------------------------------------------------------
// Pass 1: per-edge attention scores via V_WMMA_F32_16X16X4_F32.
// One wave32 handles 16 edges for one (b, h).
//   A (16x4 f32): row m = Q row of dst[ebase+m], K-chunk of 4
//   B (4x16 f32): col n = K row of src[ebase+n], K-chunk of 4
//   D[i][i] accumulated over 16 chunks = q[dst_i] . k[src_i]
// f32 A layout:  lane<16 -> (K=kb+0 in v0, kb+1 in v1); lane>=16 -> (kb+2, kb+3)
// f32 B layout (symmetric): same K split, N = lane&15
// D diagonal: i<8 at (vgpr i, lane i); i>=8 at (vgpr i-8, lane i+16)
// ---------------------------------------------------------------------------
__global__ void sa_scores(const float* __restrict__ Q, const float* __restrict__ K,
                          const long long* __restrict__ adj,
                          float* __restrict__ qk, unsigned* __restrict__ segmax,
                          int L, long long NE, float temp) {
  const int lane = threadIdx.x & 31;
  const long long wid =
      (long long)blockIdx.x * (blockDim.x >> 5) + (threadIdx.x >> 5);
  const long long nGroups = (NE + 15) >> 4;
  if (wid >= nGroups * (long long)(BB * HH)) return;  // wave-uniform

  const int bh = (int)(wid % (BB * HH));
  const long long g = wid / (BB * HH);
  const int h = bh & (HH - 1);
  const int b = bh >> 3;
  const long long ebase = g << 4;

  const int m = lane & 15;                 // A row index == B col index
  const int koff = (lane < 16) ? 0 : 2;    // K sub-offset within chunk of 4

  long long e = ebase + m;
  if (e > NE - 1) e = NE - 1;              // tail clamp (EXEC stays full)
  const long long dst = adj[e];
  const long long src = adj[NE + e];

  const float* qrow = Q + (((long long)b * L + dst) * HH + h) * (long long)EE + koff;
  const float* krow = K + (((long long)b * L + src) * HH + h) * (long long)EE + koff;

  v8f c = {0.f, 0.f, 0.f, 0.f, 0.f, 0.f, 0.f, 0.f};
#pragma unroll
  for (int kb = 0; kb < EE; kb += 4) {
    v2f a    = *(const v2f*)(qrow + kb);
    v2f bmat = *(const v2f*)(krow + kb);
    // (neg_a, A, neg_b, B, c_mod, C, reuse_a, reuse_b)
    c = __builtin_amdgcn_wmma_f32_16x16x4_f32(false, a, false, bmat,
                                              (short)0, c, false, false);
  }

  // Diagonal extraction: lanes 0..7 own D[i][i] i=lane; lanes 24..31 own i=lane-16.
  const int di = lane & 7;
  float diag = c[0];
#pragma unroll
  for (int i = 1; i < 8; ++i)
    if (di == i) diag = c[i];

  if (lane < 8 || lane >= 24) {
    const float val = diag * temp;
    qk[(e * BB + b) * HH + h] = val;
    atomicMax(&segmax[((long long)b * L + dst) * HH + h], enc_ord(val));
  }
}

// ---------------------------------------------------------------------------
// Pass 2: ex = exp(qk - segmax[dst]);  segsum[dst] += ex  (qk overwritten by ex)
// ---------------------------------------------------------------------------
__global__ void sa_expsum(const long long* __restrict__ adj,
                          float* __restrict__ qk,
                          const unsigned* __restrict__ segmax,
                          float* __restrict__ segsum, int L, long long NE) {
  long long t = (long long)blockIdx.x * blockDim.x + threadIdx.x;
  if (t >= NE * (long long)(BB * HH)) return;
  const int h = (int)(t % HH);
  long long r = t / HH;
  const int b = (int)(r % BB);
  const long long e = r / BB;
  const long long dst = adj[e];
  const long long si = ((long long)b * L + dst) * HH + h;
  const float mx = dec_ord(segmax[si]);
  const float exv = __expf(qk[t] - mx);
  qk[t] = exv;
  atomicAdd(&segsum[si], exv);
}

// ---------------------------------------------------------------------------
// Pass 3: out[b,dst,h,:] += (ex / (segsum+eps)) * v[b,src,h,:]
// 16 threads per (e,b,h), each a coalesced float4 + 4 global f32 atomics.
// ---------------------------------------------------------------------------
__global__ void sa_scatter(const float* __restrict__ V,
                           const long long* __restrict__ adj,
                           const float* __restrict__ ex,
                           const float* __restrict__ segsum,
                           float* __restrict__ out, int L, long long NE) {
  long long t = (long long)blockIdx.x * blockDim.x + threadIdx.x;
  const long long total = NE * (long long)(BB * HH) * (EE / 4);
  if (t >= total) return;
  const int q4 = (int)(t & (EE / 4 - 1));  // which float4 of the 64-wide row
  long long r = t >> 4;
  const int h = (int)(r % HH);
  r /= HH;
  const int b = (int)(r % BB);
  const long long e = r / BB;
  const long long dst = adj[e];
  const long long src = adj[NE + e];

  const float alpha = ex[(e * BB + b) * HH + h] /
                      (segsum[((long long)b * L + dst) * HH + h] + 1e-16f);

  const float* vrow = V + (((long long)b * L + src) * HH + h) * (long long)EE + q4 * 4;
  float* orow = out + (((long long)b * L + dst) * HH + h) * (long long)EE + q4 * 4;
  v4f vv = *(const v4f*)vrow;
  atomicAdd(orow + 0, alpha * vv.x);
  atomicAdd(orow + 1, alpha * vv.y);
  atomicAdd(orow + 2, alpha * vv.z);
  atomicAdd(orow + 3, alpha * vv.w);
}

// ---------------------------------------------------------------------------
extern "C" void kernel_launch(void* const* d_in, const int* in_sizes, int n_in,
                              void* d_out, int out_size, void* d_ws, size_t ws_size,
                              hipStream_t stream) {
  const float* Q = (const float*)d_in[0];
  const float* K = (const float*)d_in[1];
  const float* V = (const float*)d_in[2];
  const long long* adj = (const long long*)d_in[3];  // int64: [0]=dst row, [1]=src row
  float* out = (float*)d_out;

  const long long NE = (long long)in_sizes[3] / 2;
  const long long qsz = (long long)in_sizes[0];          // B*L*H*E
  const int L = (int)(qsz / (long long)(BB * HH * EE));
  const float temp = 1.0f / sqrtf((float)EE);

  // workspace layout: qk/ex [NE*B*H] f32 | segmax [L*B*H] u32 | segsum [L*B*H] f32
  float* qk = (float*)d_ws;
  const size_t qkBytes = (size_t)NE * BB * HH * sizeof(float);
  unsigned* segmax = (unsigned*)((char*)d_ws + qkBytes);
  const size_t segBytes = (size_t)L * BB * HH * sizeof(unsigned);
  float* segsum = (float*)((char*)segmax + segBytes);
  const long long nSeg = (long long)L * BB * HH;

  {  // pass 0: init
    long long n = (qsz > nSeg) ? qsz : nSeg;
    int blocks = (int)((n + 255) / 256);
    sa_init<<<blocks, 256, 0, stream>>>(out, qsz, segmax, segsum, nSeg);
  }
  {  // pass 1: WMMA edge scores + segment max
    long long nGroups = (NE + 15) >> 4;
    long long waves = nGroups * (long long)(BB * HH);
    int blocks = (int)((waves + 7) / 8);  // 8 wave32 per 256-thread block
    sa_scores<<<blocks, 256, 0, stream>>>(Q, K, adj, qk, segmax, L, NE, temp);
  }
  {  // pass 2: exp + segment sum
    long long tot = NE * (long long)(BB * HH);
    int blocks = (int)((tot + 255) / 256);
    sa_expsum<<<blocks, 256, 0, stream>>>(adj, qk, segmax, segsum, L, NE);
  }
  {  // pass 3: weighted scatter-add
    long long tot = NE * (long long)(BB * HH) * (EE / 4);
    int blocks = (int)((tot + 255) / 256);
    sa_scatter<<<blocks, 256, 0, stream>>>(V, adj, qk, segsum, out, L, NE);
  }
}